// Receiver_70257075027981
// MI455X (gfx1250) — compile-verified
//
#include <hip/hip_runtime.h>
#include <math.h>

typedef __attribute__((ext_vector_type(2))) float v2f;
typedef __attribute__((ext_vector_type(8))) float v8f;

#define NEG_SLOPE 0.2f

// ---------------------------------------------------------------------------
// float atomic max via ordered integer-bits trick (works for mixed signs)
// ---------------------------------------------------------------------------
__device__ __forceinline__ void atomicMaxF(float* addr, float v) {
  if (v >= 0.0f) {
    atomicMax((int*)addr, __float_as_int(v));
  } else {
    atomicMin((unsigned int*)addr, __float_as_uint(v));
  }
}

// ---------------------------------------------------------------------------
// generic fill
// ---------------------------------------------------------------------------
__global__ void k_fill(float* __restrict__ p, float v, long n) {
  long i = (long)blockIdx.x * blockDim.x + threadIdx.x;
  if (i < n) p[i] = v;
}

// ---------------------------------------------------------------------------
// edge prep: int64 -> int32 indices, copy edge attr, segment count/sum for
// the self-loop 'mean' fill value
// ---------------------------------------------------------------------------
__global__ void k_prep_edges(const long long* __restrict__ ei,
                             const float* __restrict__ eattr, int E,
                             int* __restrict__ src32, int* __restrict__ dst32,
                             float* __restrict__ eaf,
                             float* __restrict__ cnt, float* __restrict__ asum) {
  int e = blockIdx.x * blockDim.x + threadIdx.x;
  if (e >= E) return;
  int s = (int)ei[e];
  int d = (int)ei[E + e];
  float a = eattr[e];
  src32[e] = s;
  dst32[e] = d;
  eaf[e]   = a;
  atomicAdd(&cnt[d], 1.0f);
  atomicAdd(&asum[d], a);
}

__global__ void k_self_loops(int E, int N,
                             int* __restrict__ src32, int* __restrict__ dst32,
                             float* __restrict__ eaf,
                             const float* __restrict__ cnt,
                             const float* __restrict__ asum) {
  int n = blockIdx.x * blockDim.x + threadIdx.x;
  if (n >= N) return;
  src32[E + n] = n;
  dst32[E + n] = n;
  eaf[E + n]   = asum[n] / fmaxf(cnt[n], 1.0f);
}

// ---------------------------------------------------------------------------
// conv1 linear: Fin is tiny (2), do it as a VALU kernel over N*C
// ---------------------------------------------------------------------------
__global__ void k_conv1(const float* __restrict__ x,
                        const float* __restrict__ Wl, const float* __restrict__ bl,
                        const float* __restrict__ Wr, const float* __restrict__ br,
                        float* __restrict__ xl, float* __restrict__ xr,
                        int N, int C, int Fin) {
  long i = (long)blockIdx.x * blockDim.x + threadIdx.x;
  if (i >= (long)N * C) return;
  int c = (int)(i % C);
  int n = (int)(i / C);
  float al = bl[c], ar = br[c];
  for (int f = 0; f < Fin; ++f) {
    float xv = x[(long)n * Fin + f];
    al = fmaf(xv, Wl[(long)f * C + c], al);
    ar = fmaf(xv, Wr[(long)f * C + c], ar);
  }
  xl[i] = al;
  xr[i] = ar;
}

// ---------------------------------------------------------------------------
// simple transpose: Wt[n*K + k] = W[k*Nc + n]   (used once for Wl2/Wr2)
// ---------------------------------------------------------------------------
__global__ void k_transpose(const float* __restrict__ W, float* __restrict__ Wt,
                            int K, int Nc) {
  long i = (long)blockIdx.x * blockDim.x + threadIdx.x;
  if (i >= (long)K * Nc) return;
  int k = (int)(i / Nc);
  int n = (int)(i % Nc);
  Wt[(long)n * K + k] = W[i];
}

// ---------------------------------------------------------------------------
// per-edge attention logit: one wave32 per edge, shfl-xor reduction,
// atomic segment max into mx[dst]
// ---------------------------------------------------------------------------
__global__ void k_edge_logit(const float* __restrict__ xl, const float* __restrict__ xr,
                             const float* __restrict__ eaf,
                             const int* __restrict__ src32, const int* __restrict__ dst32,
                             const float* __restrict__ We, const float* __restrict__ att,
                             float* __restrict__ logits, float* __restrict__ mx,
                             int C, int Eall) {
  int wid  = (int)(((long)blockIdx.x * blockDim.x + threadIdx.x) >> 5);
  int lane = threadIdx.x & 31;
  if (wid >= Eall) return;
  int s = src32[wid];
  int d = dst32[wid];
  float ea = eaf[wid];
  float acc = 0.0f;
  for (int c = lane; c < C; c += 32) {
    float m = xl[(long)s * C + c] + xr[(long)d * C + c] + ea * We[c];
    m = (m > 0.0f) ? m : NEG_SLOPE * m;
    acc = fmaf(m, att[c], acc);
  }
  for (int off = 16; off >= 1; off >>= 1)
    acc += __shfl_xor(acc, off, 32);
  if (lane == 0) {
    logits[wid] = acc;
    atomicMaxF(&mx[d], acc);
  }
}

// ---------------------------------------------------------------------------
// exp(logit - max) and segment denominator
// ---------------------------------------------------------------------------
__global__ void k_edge_expsum(const float* __restrict__ logits,
                              const int* __restrict__ dst32,
                              const float* __restrict__ mx,
                              float* __restrict__ exw, float* __restrict__ den,
                              int Eall) {
  int e = blockIdx.x * blockDim.x + threadIdx.x;
  if (e >= Eall) return;
  int d = dst32[e];
  float ex = __expf(logits[e] - mx[d]);
  exw[e] = ex;
  atomicAdd(&den[d], ex);
}

// ---------------------------------------------------------------------------
// scatter alpha * xl[src] into agg[dst]: one wave per edge, f32 atomics
// (agg is 25.6 MB -> L2 resident at 192 MB)
// ---------------------------------------------------------------------------
__global__ void k_edge_scatter(const float* __restrict__ xl,
                               const int* __restrict__ src32, const int* __restrict__ dst32,
                               const float* __restrict__ exw, const float* __restrict__ den,
                               float* __restrict__ agg, int C, int Eall) {
  int wid  = (int)(((long)blockIdx.x * blockDim.x + threadIdx.x) >> 5);
  int lane = threadIdx.x & 31;
  if (wid >= Eall) return;
  int s = src32[wid];
  int d = dst32[wid];
  float w = exw[wid] / den[d];
  for (int c = lane; c < C; c += 32) {
    atomicAdd(&agg[(long)d * C + c], w * xl[(long)s * C + c]);
  }
}

// ---------------------------------------------------------------------------
// h = relu(agg + bias)
// ---------------------------------------------------------------------------
__global__ void k_node_out(const float* __restrict__ agg, const float* __restrict__ bias,
                           float* __restrict__ h, int N, int C) {
  long i = (long)blockIdx.x * blockDim.x + threadIdx.x;
  if (i >= (long)N * C) return;
  int c = (int)(i % C);
  float v = agg[i] + bias[c];
  h[i] = fmaxf(v, 0.0f);
}

// ---------------------------------------------------------------------------
// f32 WMMA GEMM: D[M,Nn] = A[M,K] @ Bt[Nn,K]^T (+ bias per column).
// K is contiguous in BOTH operands -> b64 fragment loads.
// One wave computes a 16x32 output strip (two 16x16 D tiles sharing the A
// fragment: 2 WMMAs per A load, independent accumulators -> no hazard NOPs).
// VGPR layouts per CDNA5 ISA 7.12.2 (V_WMMA_F32_16X16X4_F32):
//   A 16x4 : lanes 0-15 -> (M=lane, K=k0,k0+1), lanes 16-31 -> (M=lane-16, K=k0+2,k0+3)
//   B 4x16 : vreg v, lanes 0-15 -> K=k0+v, lanes 16-31 -> K=k0+v+2 ; N = lane&15
//   D 16x16: vreg r, lanes 0-15 -> M=r, lanes 16-31 -> M=r+8 ; N = lane&15
// Requirements (hold at every call site): M % 16 == 0, Nn % 32 == 0,
// K % 4 == 0, lda/ldb even (8B-aligned rows).
// ---------------------------------------------------------------------------
__global__ void k_wmma_gemm(const float* __restrict__ A, const float* __restrict__ Bt,
                            const float* __restrict__ bias, float* __restrict__ D,
                            int M, int Nn, int K, int lda, int ldb, int ldd) {
  int lane = threadIdx.x & 31;
  int tiles_n2 = Nn >> 5;                       // pairs of 16-col tiles
  int tiles    = (M >> 4) * tiles_n2;
  int tile     = blockIdx.x * blockDim.y + threadIdx.y;
  if (tile >= tiles) return;                    // wave-uniform exit, EXEC stays full
  int tm = tile / tiles_n2, tn = tile % tiles_n2;
  int m0 = tm << 4, n0 = tn << 5;
  int half = lane >> 4;                         // 0: lanes 0-15, 1: lanes 16-31
  int r16  = lane & 15;

  const float* Arow  = A  + (long)(m0 + r16) * lda;
  const float* Brow0 = Bt + (long)(n0 + r16) * ldb;
  const float* Brow1 = Bt + (long)(n0 + 16 + r16) * ldb;

  v8f acc0 = {}, acc1 = {};
#if __has_builtin(__builtin_amdgcn_wmma_f32_16x16x4_f32)
  for (int k0 = 0; k0 < K; k0 += 4) {
    int ka = k0 + 2 * half;
    v2f a  = *(const v2f*)(Arow  + ka);         // global_load_b64
    v2f b0 = *(const v2f*)(Brow0 + ka);
    v2f b1 = *(const v2f*)(Brow1 + ka);
    acc0 = __builtin_amdgcn_wmma_f32_16x16x4_f32(false, a, false, b0,
                                                 (short)0, acc0, false, false);
    acc1 = __builtin_amdgcn_wmma_f32_16x16x4_f32(false, a, false, b1,
                                                 (short)0, acc1, false, false);
  }
#else
  // scalar fallback (should not be hit on gfx1250)
  for (int r = 0; r < 8; ++r) {
    int row = m0 + r + half * 8;
    float s0 = 0.0f, s1 = 0.0f;
    for (int k = 0; k < K; ++k) {
      float av = A[(long)row * lda + k];
      s0 = fmaf(av, Bt[(long)(n0 + r16) * ldb + k], s0);
      s1 = fmaf(av, Bt[(long)(n0 + 16 + r16) * ldb + k], s1);
    }
    acc0[r] = s0; acc1[r] = s1;
  }
#endif
  int col0 = n0 + r16;
  int col1 = n0 + 16 + r16;
  float bv0 = bias ? bias[col0] : 0.0f;
  float bv1 = bias ? bias[col1] : 0.0f;
#pragma unroll
  for (int r = 0; r < 8; ++r) {
    int row = m0 + r + half * 8;
    D[(long)row * ldd + col0] = acc0[r] + bv0;
    D[(long)row * ldd + col1] = acc1[r] + bv1;
  }
}

// ---------------------------------------------------------------------------
// softmax over axis=0 (per column b of dots[N,B]); one block per column.
// dots (102 MB) is L2-resident, so the three strided passes hit L2.
// ---------------------------------------------------------------------------
__global__ void k_col_softmax(const float* __restrict__ dots, float* __restrict__ out,
                              int N, int B) {
  __shared__ float red[256];
  int b   = blockIdx.x;
  int tid = threadIdx.x;
  float m = -INFINITY;
  for (int n = tid; n < N; n += blockDim.x)
    m = fmaxf(m, dots[(long)n * B + b]);
  red[tid] = m; __syncthreads();
  for (int s = blockDim.x >> 1; s > 0; s >>= 1) {
    if (tid < s) red[tid] = fmaxf(red[tid], red[tid + s]);
    __syncthreads();
  }
  float cm = red[0]; __syncthreads();

  float sum = 0.0f;
  for (int n = tid; n < N; n += blockDim.x)
    sum += __expf(dots[(long)n * B + b] - cm);
  red[tid] = sum; __syncthreads();
  for (int s = blockDim.x >> 1; s > 0; s >>= 1) {
    if (tid < s) red[tid] += red[tid + s];
    __syncthreads();
  }
  float inv = 1.0f / red[0];
  for (int n = tid; n < N; n += blockDim.x)
    out[(long)n * B + b] = __expf(dots[(long)n * B + b] - cm) * inv;
}

// ---------------------------------------------------------------------------
// host orchestration
// ---------------------------------------------------------------------------
static inline int cdiv(long a, long b) { return (int)((a + b - 1) / b); }

extern "C" void kernel_launch(void* const* d_in, const int* in_sizes, int n_in,
                              void* d_out, int out_size, void* d_ws, size_t ws_size,
                              hipStream_t stream) {
  (void)n_in; (void)out_size; (void)ws_size;

  const float*     message = (const float*)d_in[0];
  const float*     x       = (const float*)d_in[1];
  const long long* ei      = (const long long*)d_in[2];
  const float*     eattr   = (const float*)d_in[3];
  const float* Wl1 = (const float*)d_in[4];
  const float* bl1 = (const float*)d_in[5];
  const float* Wr1 = (const float*)d_in[6];
  const float* br1 = (const float*)d_in[7];
  const float* We1 = (const float*)d_in[8];
  const float* att1= (const float*)d_in[9];
  const float* b1  = (const float*)d_in[10];
  const float* Wl2 = (const float*)d_in[11];
  const float* bl2 = (const float*)d_in[12];
  const float* Wr2 = (const float*)d_in[13];
  const float* br2 = (const float*)d_in[14];
  const float* We2 = (const float*)d_in[15];
  const float* att2= (const float*)d_in[16];
  const float* b2  = (const float*)d_in[17];
  const float* fcW = (const float*)d_in[18];
  const float* fcb = (const float*)d_in[19];

  const int C   = in_sizes[5];             // 128
  const int H   = in_sizes[18] / C;        // 256
  const int B   = in_sizes[0] / H;         // 512
  const int Fin = in_sizes[4] / C;         // 2
  const int N   = in_sizes[1] / Fin;       // 50000
  const int E   = in_sizes[3];             // 800000
  const int Eall = E + N;                  // with self loops

  // workspace carve-up (256B aligned)
  char* w = (char*)d_ws;
  auto carve = [&](size_t bytes) -> void* {
    void* p = (void*)w;
    w += (bytes + 255) & ~(size_t)255;
    return p;
  };
  int*   src32 = (int*)  carve((size_t)Eall * 4);
  int*   dst32 = (int*)  carve((size_t)Eall * 4);
  float* eaf   = (float*)carve((size_t)Eall * 4);
  float* cnt   = (float*)carve((size_t)N * 4);
  float* asum  = (float*)carve((size_t)N * 4);
  float* xl    = (float*)carve((size_t)N * C * 4);   // reused both layers
  float* xr    = (float*)carve((size_t)N * C * 4);
  float* agg   = (float*)carve((size_t)N * C * 4);
  float* h1    = (float*)carve((size_t)N * C * 4);
  float* h2    = (float*)carve((size_t)N * C * 4);
  float* logits= (float*)carve((size_t)Eall * 4);
  float* exw   = (float*)carve((size_t)Eall * 4);
  float* mx    = (float*)carve((size_t)N * 4);
  float* den   = (float*)carve((size_t)N * 4);
  float* me    = (float*)carve((size_t)B * C * 4);
  float* Wlt   = (float*)carve((size_t)C * C * 4);   // Wl2 transposed
  float* Wrt   = (float*)carve((size_t)C * C * 4);   // Wr2 transposed
  float* dots  = (float*)carve((size_t)N * B * 4);

  const int T = 256;

  // --- self-loop mean edge attr ---
  k_fill<<<cdiv(N, T), T, 0, stream>>>(cnt, 0.0f, N);
  k_fill<<<cdiv(N, T), T, 0, stream>>>(asum, 0.0f, N);
  k_prep_edges<<<cdiv(E, T), T, 0, stream>>>(ei, eattr, E, src32, dst32, eaf, cnt, asum);
  k_self_loops<<<cdiv(N, T), T, 0, stream>>>(E, N, src32, dst32, eaf, cnt, asum);

  // --- conv1 linears (Fin=2, VALU) ---
  k_conv1<<<cdiv((long)N * C, T), T, 0, stream>>>(x, Wl1, bl1, Wr1, br1, xl, xr, N, C, Fin);

  // --- transpose conv2 weights so GEMM K-dim is contiguous in both operands ---
  k_transpose<<<cdiv((long)C * C, T), T, 0, stream>>>(Wl2, Wlt, C, C);
  k_transpose<<<cdiv((long)C * C, T), T, 0, stream>>>(Wr2, Wrt, C, C);

  // --- layer 1 edge phase ---
  k_fill<<<cdiv(N, T), T, 0, stream>>>(mx, -INFINITY, N);
  k_fill<<<cdiv(N, T), T, 0, stream>>>(den, 0.0f, N);
  k_fill<<<cdiv((long)N * C, T), T, 0, stream>>>(agg, 0.0f, (long)N * C);
  k_edge_logit  <<<cdiv((long)Eall * 32, T), T, 0, stream>>>(xl, xr, eaf, src32, dst32,
                                                             We1, att1, logits, mx, C, Eall);
  k_edge_expsum <<<cdiv(Eall, T), T, 0, stream>>>(logits, dst32, mx, exw, den, Eall);
  k_edge_scatter<<<cdiv((long)Eall * 32, T), T, 0, stream>>>(xl, src32, dst32, exw, den,
                                                             agg, C, Eall);
  k_node_out<<<cdiv((long)N * C, T), T, 0, stream>>>(agg, b1, h1, N, C);

  // --- layer 2 linears: WMMA GEMMs (M=N=50000, Nn=K=C=128) ---
  {
    dim3 blk(32, 8);
    int tiles = (N >> 4) * (C >> 5);
    int grid  = cdiv(tiles, 8);
    k_wmma_gemm<<<grid, blk, 0, stream>>>(h1, Wlt, bl2, xl, N, C, C, C, C, C);
    k_wmma_gemm<<<grid, blk, 0, stream>>>(h1, Wrt, br2, xr, N, C, C, C, C, C);
  }

  // --- layer 2 edge phase ---
  k_fill<<<cdiv(N, T), T, 0, stream>>>(mx, -INFINITY, N);
  k_fill<<<cdiv(N, T), T, 0, stream>>>(den, 0.0f, N);
  k_fill<<<cdiv((long)N * C, T), T, 0, stream>>>(agg, 0.0f, (long)N * C);
  k_edge_logit  <<<cdiv((long)Eall * 32, T), T, 0, stream>>>(xl, xr, eaf, src32, dst32,
                                                             We2, att2, logits, mx, C, Eall);
  k_edge_expsum <<<cdiv(Eall, T), T, 0, stream>>>(logits, dst32, mx, exw, den, Eall);
  k_edge_scatter<<<cdiv((long)Eall * 32, T), T, 0, stream>>>(xl, src32, dst32, exw, den,
                                                             agg, C, Eall);
  k_node_out<<<cdiv((long)N * C, T), T, 0, stream>>>(agg, b2, h2, N, C);

  // --- me = message @ fcW^T + fcb : M=B(512), Nn=C(128), K=H(256)
  //     fcW [C,H] row-major is already Bt (Bt[c,h] = fcW[c*H+h])
  {
    dim3 blk(32, 8);
    int tiles = (B >> 4) * (C >> 5);
    k_wmma_gemm<<<cdiv(tiles, 8), blk, 0, stream>>>(message, fcW, fcb, me,
                                                    B, C, H, H, H, C);
  }

  // --- dots = h2 @ me^T : M=N, Nn=B(512), K=C(128)
  //     me [B,C] row-major is already Bt (Bt[b,c] = me[b*C+c]); no bias
  {
    dim3 blk(32, 8);
    int tiles = (N >> 4) * (B >> 5);
    k_wmma_gemm<<<cdiv(tiles, 8), blk, 0, stream>>>(h2, me, nullptr, dots,
                                                    N, B, C, C, C, B);
  }

  // --- softmax over nodes per column -> d_out [N,B] ---
  k_col_softmax<<<B, 256, 0, stream>>>(dots, (float*)d_out, N, B);
}